// GAT_22892175688227
// MI455X (gfx1250) — compile-verified
//
#include <hip/hip_runtime.h>
#include <hip/hip_bf16.h>

typedef __attribute__((ext_vector_type(2))) float v2f;
typedef __attribute__((ext_vector_type(8))) float v8f;

#define HDIM 128
#define CTPW 4   // column tiles per wave in the main GEMM

// ---------------------------------------------------------------------------
// Transpose W[rows x cols] -> Wt[colsPad x rows], zero-padding extra columns.
// Tiny one-off (<=64KB); amortized over 1250 row tiles.
// ---------------------------------------------------------------------------
__global__ __launch_bounds__(256) void gat_transpose(
    const float* __restrict__ W, float* __restrict__ Wt,
    int rows, int cols, int colsPad)
{
    int i = blockIdx.x * blockDim.x + threadIdx.x;
    int total = colsPad * rows;
    if (i >= total) return;
    int c = i / rows;
    int r = i - c * rows;
    Wt[(size_t)c * rows + r] = (c < cols) ? W[(size_t)r * cols + c] : 0.f;
}

// ---------------------------------------------------------------------------
// GEMM: H[N x 128] = X[N x 128] @ W[128 x 128] using V_WMMA_F32_16X16X4_F32.
// One wave32 per (16-row strip, 4 column tiles). W pre-transposed so both A
// and B fragments are contiguous b64 loads.
// A (16x4 f32): lanes 0-15 -> M=lane, VGPR{0,1}=K{0,1}; lanes 16-31 -> K{2,3}.
// B (4x16 f32): VGPR0: lanes0-15 = row K0, lanes16-31 = row K2; VGPR1: K1/K3.
// C/D (16x16 f32): VGPR r: lanes0-15 -> (M=r, N=lane); lanes16-31 -> (M=r+8).
// ---------------------------------------------------------------------------
__global__ __launch_bounds__(256) void gat_gemm_strip(
    const float* __restrict__ X, const float* __restrict__ Wt,
    float* __restrict__ H, int nRowTiles)
{
    int wid  = (blockIdx.x * blockDim.x + threadIdx.x) >> 5;
    int lane = threadIdx.x & 31;
    int nHalves = HDIM / (CTPW * 16);          // 2
    int nW = nRowTiles * nHalves;
    if (wid >= nW) return;
    int rt   = wid >> 1;
    int half = wid & 1;
    int row0 = rt * 16;
    int laneM = lane & 15;
    int kOff  = (lane >> 4) * 2;               // 0 / 2

    const float* xrow = X  + (size_t)(row0 + laneM) * HDIM + kOff;
    const float* wrow = Wt + (size_t)(half * (CTPW * 16) + laneM) * HDIM + kOff;

    v8f c[CTPW] = {};
    #pragma unroll 4
    for (int k = 0; k < HDIM; k += 4) {
        v2f a = *(const v2f*)(xrow + k);
        #pragma unroll
        for (int t = 0; t < CTPW; ++t) {
            v2f b = *(const v2f*)(wrow + (size_t)t * 16 * HDIM + k);
            c[t] = __builtin_amdgcn_wmma_f32_16x16x4_f32(
                       false, a, false, b, (short)0, c[t], false, false);
        }
    }

    int mBase = (lane >> 4) * 8;
    #pragma unroll
    for (int t = 0; t < CTPW; ++t) {
        float* o = H + (size_t)row0 * HDIM + (half * CTPW + t) * 16 + laneM;
        #pragma unroll
        for (int r = 0; r < 8; ++r)
            o[(size_t)(mBase + r) * HDIM] = c[t][r];
    }
}

// ---------------------------------------------------------------------------
// Classifier: Out[N x 40] = X[N x 128] @ Wl[128 x 40] + bl.
// Wlt is the zero-padded transpose [48 x 128]; one wave per 16-row strip
// owns all 3 column tiles; column guard only at the store.
// ---------------------------------------------------------------------------
__global__ __launch_bounds__(256) void gat_gemm_cls(
    const float* __restrict__ X, const float* __restrict__ Wlt,
    const float* __restrict__ bl, float* __restrict__ Out,
    int nRowTiles, int nCols)
{
    int wid  = (blockIdx.x * blockDim.x + threadIdx.x) >> 5;
    int lane = threadIdx.x & 31;
    if (wid >= nRowTiles) return;
    int row0 = wid * 16;
    int laneM = lane & 15;
    int kOff  = (lane >> 4) * 2;

    const float* xrow = X   + (size_t)(row0 + laneM) * HDIM + kOff;
    const float* wrow = Wlt + (size_t)laneM * HDIM + kOff;

    v8f c[3] = {};
    #pragma unroll 4
    for (int k = 0; k < HDIM; k += 4) {
        v2f a = *(const v2f*)(xrow + k);
        #pragma unroll
        for (int t = 0; t < 3; ++t) {
            v2f b = *(const v2f*)(wrow + (size_t)t * 16 * HDIM + k);
            c[t] = __builtin_amdgcn_wmma_f32_16x16x4_f32(
                       false, a, false, b, (short)0, c[t], false, false);
        }
    }

    int mBase = (lane >> 4) * 8;
    #pragma unroll
    for (int t = 0; t < 3; ++t) {
        int col = t * 16 + laneM;
        if (col >= nCols) continue;
        float bv = bl[col];
        #pragma unroll
        for (int r = 0; r < 8; ++r)
            Out[(size_t)(row0 + mBase + r) * nCols + col] = c[t][r] + bv;
    }
}

// ---------------------------------------------------------------------------
// Per-node attention logits: aS[n] = h[n,:]·a_src, aD[n] = h[n,:]·a_dst.
// One wave per node, wave32 shfl_xor reduction.
// ---------------------------------------------------------------------------
__global__ __launch_bounds__(256) void gat_alpha(
    const float* __restrict__ H, const float* __restrict__ asrc,
    const float* __restrict__ adst, float* __restrict__ aS,
    float* __restrict__ aD, int N)
{
    int w    = (blockIdx.x * blockDim.x + threadIdx.x) >> 5;
    int lane = threadIdx.x & 31;
    if (w >= N) return;
    const float* h = H + (size_t)w * HDIM;
    float s = 0.f, d = 0.f;
    #pragma unroll
    for (int j = 0; j < 4; ++j) {
        int f = lane + 32 * j;
        float hv = h[f];
        s += hv * asrc[f];
        d += hv * adst[f];
    }
    #pragma unroll
    for (int off = 16; off > 0; off >>= 1) {
        s += __shfl_xor(s, off, 32);
        d += __shfl_xor(d, off, 32);
    }
    if (lane == 0) { aS[w] = s; aD[w] = d; }
}

// Order-preserving float<->uint mapping for atomic segment-max.
__device__ __forceinline__ unsigned f2ord(float f) {
    unsigned b = __float_as_uint(f);
    return (b & 0x80000000u) ? ~b : (b | 0x80000000u);
}
__device__ __forceinline__ float ord2f(unsigned u) {
    unsigned b = (u & 0x80000000u) ? (u & 0x7FFFFFFFu) : ~u;
    return __uint_as_float(b);
}

// init: agg[N*128] = 0, segmax = enc(-inf), den = 0
__global__ __launch_bounds__(256) void gat_init(
    unsigned* __restrict__ segmax, float* __restrict__ den,
    float* __restrict__ agg, int N)
{
    int i = blockIdx.x * blockDim.x + threadIdx.x;
    int total = N * HDIM;
    if (i < total) agg[i] = 0.f;
    if (i < N) {
        segmax[i] = 0x007FFFFFu;   // enc(-inf)
        den[i]    = 0.f;
    }
}

// Pass 1: e = leaky_relu(aS[src] + aD[dst]); segment max over dst.
__global__ __launch_bounds__(256) void gat_edge1(
    const int* __restrict__ ei, int E, int N,
    const float* __restrict__ aS, const float* __restrict__ aD,
    float* __restrict__ eVal, unsigned* __restrict__ segmax)
{
    int i = blockIdx.x * blockDim.x + threadIdx.x;
    int ET = E + N;
    if (i >= ET) return;
    int s, d;
    if (i < E) { s = ei[i]; d = ei[E + i]; } else { s = d = i - E; }
    float e = aS[s] + aD[d];
    e = (e >= 0.f) ? e : 0.2f * e;
    eVal[i] = e;
    atomicMax(&segmax[d], f2ord(e));
}

// Pass 2: ex = exp(e - max[dst]); segment sum over dst.
__global__ __launch_bounds__(256) void gat_edge2(
    const int* __restrict__ ei, int E, int N,
    float* __restrict__ eVal, const unsigned* __restrict__ segmax,
    float* __restrict__ den)
{
    int i = blockIdx.x * blockDim.x + threadIdx.x;
    int ET = E + N;
    if (i >= ET) return;
    int d = (i < E) ? ei[E + i] : (i - E);
    float ex = expf(eVal[i] - ord2f(segmax[d]));
    eVal[i] = ex;
    atomicAdd(&den[d], ex);
}

// Pass 3: agg[dst,:] += h[src,:] * (ex / den[dst]).  One wave per edge,
// 4 features per lane (strided for coalescing). h and agg are L2-resident.
__global__ __launch_bounds__(256) void gat_edge3(
    const int* __restrict__ ei, int E, int N,
    const float* __restrict__ eVal, const float* __restrict__ den,
    const float* __restrict__ H, float* __restrict__ agg)
{
    int w    = (blockIdx.x * blockDim.x + threadIdx.x) >> 5;
    int lane = threadIdx.x & 31;
    int ET = E + N;
    if (w >= ET) return;
    int s, d;
    if (w < E) { s = ei[w]; d = ei[E + w]; } else { s = d = w - E; }
    float alpha = eVal[w] / den[d];
    const float* h = H + (size_t)s * HDIM;
    float* o = agg + (size_t)d * HDIM;
    #pragma unroll
    for (int j = 0; j < 4; ++j) {
        int f = lane + 32 * j;
        atomicAdd(&o[f], h[f] * alpha);
    }
}

// act = relu(agg + bias), in place.
__global__ __launch_bounds__(256) void gat_bias_relu(
    float* __restrict__ agg, const float* __restrict__ b, int total)
{
    int i = blockIdx.x * blockDim.x + threadIdx.x;
    if (i >= total) return;
    float v = agg[i] + b[i & (HDIM - 1)];
    agg[i] = (v > 0.f) ? v : 0.f;
}

// ---------------------------------------------------------------------------
extern "C" void kernel_launch(void* const* d_in, const int* in_sizes, int n_in,
                              void* d_out, int out_size, void* d_ws, size_t ws_size,
                              hipStream_t stream)
{
    const float* x   = (const float*)d_in[0];
    const int*   ei  = (const int*)d_in[1];
    const float* W1  = (const float*)d_in[2];
    const float* a1s = (const float*)d_in[3];
    const float* a1d = (const float*)d_in[4];
    const float* b1  = (const float*)d_in[5];
    const float* W2  = (const float*)d_in[6];
    const float* a2s = (const float*)d_in[7];
    const float* a2d = (const float*)d_in[8];
    const float* b2  = (const float*)d_in[9];
    const float* Wl  = (const float*)d_in[10];
    const float* bl  = (const float*)d_in[11];

    const int N  = in_sizes[0] / HDIM;   // 20000
    const int E  = in_sizes[1] / 2;      // 640000
    const int C  = in_sizes[11];         // 40
    const int CP = (C + 15) & ~15;       // 48
    const int ET = E + N;

    // workspace layout (256B aligned slabs)
    char* ws = (char*)d_ws;
    auto align256 = [](size_t v) { return (v + 255) & ~(size_t)255; };
    size_t off = 0;
    float*    bufH  = (float*)(ws + off);  off += align256((size_t)N * HDIM * 4);
    float*    bufA  = (float*)(ws + off);  off += align256((size_t)N * HDIM * 4); // agg / act
    float*    Wt    = (float*)(ws + off);  off += align256((size_t)HDIM * HDIM * 4);
    float*    Wlt   = (float*)(ws + off);  off += align256((size_t)CP * HDIM * 4);
    float*    aS    = (float*)(ws + off);  off += align256((size_t)N * 4);
    float*    aD    = (float*)(ws + off);  off += align256((size_t)N * 4);
    unsigned* smax  = (unsigned*)(ws + off); off += align256((size_t)N * 4);
    float*    den   = (float*)(ws + off);  off += align256((size_t)N * 4);
    float*    eVal  = (float*)(ws + off);  off += align256((size_t)ET * 4);
    (void)ws_size;

    const int nRowTiles = N / 16;        // 1250 (N divisible by 16)
    const int BLK = 256;
    dim3 blk(BLK);

    dim3 trGrid((HDIM * HDIM + BLK - 1) / BLK);
    dim3 gemmGrid((nRowTiles * 2 + 7) / 8);                    // 2 halves, 8 waves/blk
    dim3 waveGridN((N + 7) / 8);                               // wave per node
    dim3 elemGridN((N * HDIM + BLK - 1) / BLK);
    dim3 edgeGrid((ET + BLK - 1) / BLK);
    dim3 edgeWGrid((ET + 7) / 8);                              // wave per edge

    auto run_layer = [&](const float* act_in, const float* W,
                         const float* as_, const float* ad_, const float* bias) {
        gat_transpose<<<trGrid, blk, 0, stream>>>(W, Wt, HDIM, HDIM, HDIM);
        gat_gemm_strip<<<gemmGrid, blk, 0, stream>>>(act_in, Wt, bufH, nRowTiles);
        gat_alpha<<<waveGridN, blk, 0, stream>>>(bufH, as_, ad_, aS, aD, N);
        gat_init<<<elemGridN, blk, 0, stream>>>(smax, den, bufA, N);
        gat_edge1<<<edgeGrid, blk, 0, stream>>>(ei, E, N, aS, aD, eVal, smax);
        gat_edge2<<<edgeGrid, blk, 0, stream>>>(ei, E, N, eVal, smax, den);
        gat_edge3<<<edgeWGrid, blk, 0, stream>>>(ei, E, N, eVal, den, bufH, bufA);
        gat_bias_relu<<<elemGridN, blk, 0, stream>>>(bufA, bias, N * HDIM);
    };

    // Layer 1: input x -> bufA holds relu(gat_conv1(x))
    run_layer(x, W1, a1s, a1d, b1);
    // Layer 2: input bufA -> bufH = bufA @ W2; aggregation back into bufA
    run_layer(bufA, W2, a2s, a2d, b2);

    // Classifier: d_out = bufA @ Wl + bl  (Wl transposed + zero-padded to 48)
    dim3 trlGrid((CP * HDIM + BLK - 1) / BLK);
    gat_transpose<<<trlGrid, blk, 0, stream>>>(Wl, Wlt, HDIM, C, CP);
    dim3 clsGrid((nRowTiles + 7) / 8);
    gat_gemm_cls<<<clsGrid, blk, 0, stream>>>(bufA, Wlt, bl, (float*)d_out,
                                              nRowTiles, C);
}